// XEC_Inpainter_85633057947837
// MI455X (gfx1250) — compile-verified
//
#include <hip/hip_runtime.h>
#include <hip/hip_bf16.h>

#define NS 4760          // N_SENSORS
#define BB 16            // batch
#define NCHUNK 595       // NS / 8

typedef __attribute__((ext_vector_type(2))) float v2f;
typedef __attribute__((ext_vector_type(8))) float v8f;

__device__ static inline v8f wmma4(v2f a, v2f b, v8f c) {
    // V_WMMA_F32_16X16X4_F32 : D(16x16 f32) = A(16x4) * B(4x16) + C
    return __builtin_amdgcn_wmma_f32_16x16x4_f32(
        false, a, false, b, (short)0, c, false, false);
}

// ---------------------------------------------------------------------------
// K1: per-sensor static projections.
// query[n] = [pos_embed[n] (96), face_emb[face_ids[n]] (32)]  (128 dims)
// q_local    = query @ W_ql  + b_ql          (cols 0..63   of combined 256)
// q_g        = query @ W_qg  + b_qg          (cols 64..191)
// static_nbr = query @ W_nbr[2:130] + b_nbr  (cols 192..255)
// One block = one 16-row tile; 8 waves x 2 col-tiles each = 16 col tiles.
// ---------------------------------------------------------------------------
__global__ void k1_static(const float* __restrict__ pos_embed,
                          const float* __restrict__ face_emb,
                          const int*   __restrict__ face_ids,
                          const float* __restrict__ W_ql, const float* __restrict__ b_ql,
                          const float* __restrict__ W_qg, const float* __restrict__ b_qg,
                          const float* __restrict__ W_nbr, const float* __restrict__ b_nbr,
                          float* __restrict__ q_local,
                          float* __restrict__ q_g,
                          float* __restrict__ static_nbr) {
    __shared__ float As[16][132];
    const int rt  = blockIdx.x;
    const int tid = threadIdx.x;
    for (int i = tid; i < 16 * 128; i += 256) {
        int m = i >> 7, c = i & 127;
        int n = rt * 16 + m; if (n >= NS) n = NS - 1;
        As[m][c] = (c < 96) ? pos_embed[n * 96 + c]
                            : face_emb[face_ids[n] * 32 + (c - 96)];
    }
    __syncthreads();

    const int wave = tid >> 5, lane = tid & 31;
    const int mrow = lane & 15;
    const int kb   = (lane >> 4) * 2;

    for (int sub = 0; sub < 2; ++sub) {
        const int ct = wave * 2 + sub;      // 0..15
        const int g0 = ct * 16;             // combined column base
        const float *Wp, *bp; float* outp; int ldw, ldo, colbase;
        if (g0 < 64)        { Wp = W_ql;        bp = b_ql;  outp = q_local;    ldw = 64;  ldo = 64;  colbase = g0; }
        else if (g0 < 192)  { Wp = W_qg;        bp = b_qg;  outp = q_g;        ldw = 128; ldo = 128; colbase = g0 - 64; }
        else                { Wp = W_nbr + 128; bp = b_nbr; outp = static_nbr; ldw = 64;  ldo = 64;  colbase = g0 - 192; }
        const int ncol = colbase + mrow;
        v8f acc = {};
        for (int k0 = 0; k0 < 128; k0 += 4) {
            v2f a, b;
            a.x = As[mrow][k0 + kb];
            a.y = As[mrow][k0 + kb + 1];
            b.x = Wp[(k0 + kb) * ldw + ncol];
            b.y = Wp[(k0 + kb + 1) * ldw + ncol];
            acc = wmma4(a, b, acc);
        }
        const float bias = bp[ncol];
        #pragma unroll
        for (int r = 0; r < 8; ++r) {
            int m = r + ((lane >> 4) << 3);
            int n = rt * 16 + m;
            if (n < NS) outp[n * ldo + ncol] = acc[r] + bias;
        }
    }
}

// ---------------------------------------------------------------------------
// K2a: latent_kv[b,t] = latent_seq[b,t] @ W_lat + b_lat + (face_emb[t]@W_lf + b_lf)
// 96 tokens, 128 cols. One block per token, thread per column.
// ---------------------------------------------------------------------------
__global__ void k2_latent(const float* __restrict__ latent_seq,
                          const float* __restrict__ W_lat, const float* __restrict__ b_lat,
                          const float* __restrict__ face_emb,
                          const float* __restrict__ W_lf, const float* __restrict__ b_lf,
                          float* __restrict__ latent_kv) {
    __shared__ float row[1024];
    const int tok = blockIdx.x;     // b*6 + t
    const int t   = tok % 6;
    const int c   = threadIdx.x;    // 0..127
    for (int i = c; i < 1024; i += 128) row[i] = latent_seq[tok * 1024 + i];
    __syncthreads();
    float s = b_lat[c];
    for (int k = 0; k < 1024; ++k) s += row[k] * W_lat[k * 128 + c];
    float fb = b_lf[c];
    for (int k = 0; k < 32; ++k) fb += face_emb[t * 32 + k] * W_lf[k * 128 + c];
    latent_kv[tok * 128 + c] = s + fb;
}

// K2b: k_g / v_g = latent_kv @ W_k / W_v (+bias)
__global__ void k2b_kv(const float* __restrict__ latent_kv,
                       const float* __restrict__ W_k, const float* __restrict__ b_k,
                       const float* __restrict__ W_v, const float* __restrict__ b_v,
                       float* __restrict__ k_g, float* __restrict__ v_g) {
    __shared__ float row[128];
    const int tok = blockIdx.x, c = threadIdx.x;
    row[c] = latent_kv[tok * 128 + c];
    __syncthreads();
    float sk = b_k[c], sv = b_v[c];
    for (int k = 0; k < 128; ++k) {
        float r = row[k];
        sk += r * W_k[k * 128 + c];
        sv += r * W_v[k * 128 + c];
    }
    k_g[tok * 128 + c] = sk;
    v_g[tok * 128 + c] = sv;
}

// ---------------------------------------------------------------------------
// K3: dyn[b,j,h] = static_nbr[j,h] + x_flat[b,j,0]*W_nbr[0,h] + x_flat[b,j,1]*W_nbr[1,h]
// (this IS nbr_feat for neighbor j — the 20-GFLOP reference GEMM collapsed)
// ---------------------------------------------------------------------------
__global__ void k3_dyn(const float* __restrict__ x_flat,
                       const float* __restrict__ static_nbr,
                       const float* __restrict__ W_nbr,
                       float* __restrict__ dyn) {
    const long idx = (long)blockIdx.x * 256 + threadIdx.x;   // over B*NS*64
    if (idx >= (long)BB * NS * 64) return;
    const int  h  = (int)(idx & 63);
    const long bj = idx >> 6;                                // b*NS + j
    const int  j  = (int)(bj % NS);
    const float x0 = x_flat[bj * 2 + 0], x1 = x_flat[bj * 2 + 1];
    dyn[idx] = static_nbr[j * 64 + h] + x0 * W_nbr[h] + x1 * W_nbr[64 + h];
}

// ---------------------------------------------------------------------------
// K4: local attention. One wave per (b,n); lane owns h and h+32.
// ---------------------------------------------------------------------------
__global__ void k4_local(const float* __restrict__ dyn,
                         const float* __restrict__ q_local,
                         const int*   __restrict__ knn,
                         const int*   __restrict__ mask,
                         float* __restrict__ local_feat) {
    const int b = blockIdx.x / NCHUNK, chunk = blockIdx.x % NCHUNK;
    const int wave = threadIdx.x >> 5, lane = threadIdx.x & 31;
    const int n = chunk * 8 + wave;
    const long bbase = (long)b * NS;
    const float q0 = q_local[n * 64 + lane];
    const float q1 = q_local[n * 64 + 32 + lane];
    float f0[16], f1[16], lg[16];
    #pragma unroll
    for (int k = 0; k < 16; ++k) {
        const int j = knn[n * 16 + k];
        const float a0 = dyn[(bbase + j) * 64 + lane];
        const float a1 = dyn[(bbase + j) * 64 + 32 + lane];
        f0[k] = a0; f1[k] = a1;
        float p = a0 * q0 + a1 * q1;
        for (int off = 16; off > 0; off >>= 1) p += __shfl_xor(p, off, 32);
        lg[k] = (mask[bbase + j] != 0) ? -10000.0f : p * 0.125f;   // HID^-0.5
    }
    float mx = lg[0];
    #pragma unroll
    for (int k = 1; k < 16; ++k) mx = fmaxf(mx, lg[k]);
    float s = 0.f, w[16];
    #pragma unroll
    for (int k = 0; k < 16; ++k) { w[k] = __expf(lg[k] - mx); s += w[k]; }
    const float inv = 1.0f / s;
    float o0 = 0.f, o1 = 0.f;
    #pragma unroll
    for (int k = 0; k < 16; ++k) { const float ww = w[k] * inv; o0 += ww * f0[k]; o1 += ww * f1[k]; }
    local_feat[(bbase + n) * 64 + lane]      = o0;
    local_feat[(bbase + n) * 64 + 32 + lane] = o1;
}

// ---------------------------------------------------------------------------
// K5: global attention over 6 latent tokens, 4 heads x 32. Wave per (b,n).
// ---------------------------------------------------------------------------
__global__ void k5_global(const float* __restrict__ q_g,
                          const float* __restrict__ k_g,
                          const float* __restrict__ v_g,
                          float* __restrict__ gpre) {
    __shared__ float kg[768], vg[768];
    const int b = blockIdx.x / NCHUNK, chunk = blockIdx.x % NCHUNK;
    const int tid = threadIdx.x;
    for (int i = tid; i < 768; i += 256) { kg[i] = k_g[b * 768 + i]; vg[i] = v_g[b * 768 + i]; }
    __syncthreads();
    const int wave = tid >> 5, lane = tid & 31;
    const int n = chunk * 8 + wave;
    float qh[4];
    #pragma unroll
    for (int h = 0; h < 4; ++h) qh[h] = q_g[n * 128 + h * 32 + lane];
    float lg[4][6];
    const float sc = 0.17677669529663687f;   // HEAD_DIM^-0.5
    #pragma unroll
    for (int t = 0; t < 6; ++t)
        #pragma unroll
        for (int h = 0; h < 4; ++h) {
            float p = qh[h] * kg[t * 128 + h * 32 + lane];
            for (int off = 16; off > 0; off >>= 1) p += __shfl_xor(p, off, 32);
            lg[h][t] = p * sc;
        }
    const long obase = ((long)b * NS + n) * 128;
    #pragma unroll
    for (int h = 0; h < 4; ++h) {
        float mx = lg[h][0];
        #pragma unroll
        for (int t = 1; t < 6; ++t) mx = fmaxf(mx, lg[h][t]);
        float s = 0.f, w[6];
        #pragma unroll
        for (int t = 0; t < 6; ++t) { w[t] = __expf(lg[h][t] - mx); s += w[t]; }
        const float inv = 1.0f / s;
        float o = 0.f;
        #pragma unroll
        for (int t = 0; t < 6; ++t) o += w[t] * inv * vg[t * 128 + h * 32 + lane];
        gpre[obase + h * 32 + lane] = o;
    }
}

// ---------------------------------------------------------------------------
// K6: gbuf <- gbuf @ W_go + b_go, IN PLACE (tile staged in LDS before stores).
// 76160 rows = 4760 tiles; 8 waves, one 16-col tile each.
// ---------------------------------------------------------------------------
__global__ void k6_go(const float* __restrict__ W_go,
                      const float* __restrict__ b_go,
                      float* __restrict__ gbuf) {
    __shared__ float As[16][132];
    const int rt = blockIdx.x, tid = threadIdx.x;
    __builtin_prefetch(W_go, 0, 3);
    for (int i = tid; i < 16 * 128; i += 256) {
        int m = i >> 7, c = i & 127;
        As[m][c] = gbuf[(long)(rt * 16 + m) * 128 + c];
    }
    __syncthreads();
    const int wave = tid >> 5, lane = tid & 31;
    const int mrow = lane & 15, kb = (lane >> 4) * 2;
    const int col = wave * 16 + mrow;
    v8f acc = {};
    for (int k0 = 0; k0 < 128; k0 += 4) {
        v2f a, b;
        a.x = As[mrow][k0 + kb];     a.y = As[mrow][k0 + kb + 1];
        b.x = W_go[(k0 + kb) * 128 + col];
        b.y = W_go[(k0 + kb + 1) * 128 + col];
        acc = wmma4(a, b, acc);
    }
    const float bias = b_go[col];
    #pragma unroll
    for (int r = 0; r < 8; ++r) {
        int m = r + ((lane >> 4) << 3);
        gbuf[(long)(rt * 16 + m) * 128 + col] = acc[r] + bias;
    }
}

// ---------------------------------------------------------------------------
// K8: fused LayerNorm(concat(local,g)) -> @W_m1 + b_m1 -> exact GELU -> h
// 128 threads (4 waves, one 16-col tile each), K = 192.
// ---------------------------------------------------------------------------
__global__ void k8_mlp1(const float* __restrict__ local_feat,
                        const float* __restrict__ gpost,
                        const float* __restrict__ ln_g, const float* __restrict__ ln_b,
                        const float* __restrict__ W_m1, const float* __restrict__ b_m1,
                        float* __restrict__ hbuf) {
    __shared__ float As[16][196];
    __shared__ float mu_s[16], rs_s[16];
    const int rt = blockIdx.x, tid = threadIdx.x;
    for (int i = tid; i < 16 * 192; i += 128) {
        const int m = i / 192, c = i % 192;
        const long row = (long)rt * 16 + m;
        As[m][c] = (c < 64) ? local_feat[row * 64 + c] : gpost[row * 128 + (c - 64)];
    }
    __syncthreads();
    if (tid < 16) {
        float s = 0.f, s2 = 0.f;
        for (int c = 0; c < 192; ++c) { const float v = As[tid][c]; s += v; s2 += v * v; }
        const float mu  = s * (1.0f / 192.0f);
        const float var = s2 * (1.0f / 192.0f) - mu * mu;
        mu_s[tid] = mu;
        rs_s[tid] = rsqrtf(var + 1e-5f);
    }
    __syncthreads();
    for (int i = tid; i < 16 * 192; i += 128) {
        const int m = i / 192, c = i % 192;
        As[m][c] = (As[m][c] - mu_s[m]) * rs_s[m] * ln_g[c] + ln_b[c];
    }
    __syncthreads();
    const int wave = tid >> 5, lane = tid & 31;
    const int mrow = lane & 15, kb = (lane >> 4) * 2;
    const int col = wave * 16 + mrow;   // 0..63
    v8f acc = {};
    for (int k0 = 0; k0 < 192; k0 += 4) {
        v2f a, b;
        a.x = As[mrow][k0 + kb];     a.y = As[mrow][k0 + kb + 1];
        b.x = W_m1[(k0 + kb) * 64 + col];
        b.y = W_m1[(k0 + kb + 1) * 64 + col];
        acc = wmma4(a, b, acc);
    }
    const float bias = b_m1[col];
    #pragma unroll
    for (int r = 0; r < 8; ++r) {
        const int m = r + ((lane >> 4) << 3);
        const float x = acc[r] + bias;
        const float g = 0.5f * x * (1.0f + erff(x * 0.70710678118654752f));
        hbuf[((long)rt * 16 + m) * 64 + col] = g;
    }
}

// K9: preds = (h @ W_m2 + b_m2) * (mask != 0)
__global__ void k9_out(const float* __restrict__ hbuf,
                       const float* __restrict__ W_m2, const float* __restrict__ b_m2,
                       const int* __restrict__ mask,
                       float* __restrict__ out) {
    const long idx = (long)blockIdx.x * 256 + threadIdx.x;   // b*NS + n
    if (idx >= (long)BB * NS) return;
    float s = b_m2[0];
    for (int c = 0; c < 64; ++c) s += hbuf[idx * 64 + c] * W_m2[c];
    out[idx] = (mask[idx] != 0) ? s : 0.0f;
}

// ---------------------------------------------------------------------------
extern "C" void kernel_launch(void* const* d_in, const int* in_sizes, int n_in,
                              void* d_out, int out_size, void* d_ws, size_t ws_size,
                              hipStream_t stream) {
    (void)in_sizes; (void)n_in; (void)out_size; (void)ws_size;
    const float* x_flat     = (const float*)d_in[0];
    const float* latent_seq = (const float*)d_in[1];
    const float* pos_embed  = (const float*)d_in[2];
    const float* face_emb   = (const float*)d_in[3];
    const float* W_nbr = (const float*)d_in[4];  const float* b_nbr = (const float*)d_in[5];
    const float* W_ql  = (const float*)d_in[6];  const float* b_ql  = (const float*)d_in[7];
    const float* W_lat = (const float*)d_in[8];  const float* b_lat = (const float*)d_in[9];
    const float* W_lf  = (const float*)d_in[10]; const float* b_lf  = (const float*)d_in[11];
    const float* W_qg  = (const float*)d_in[12]; const float* b_qg  = (const float*)d_in[13];
    const float* W_k   = (const float*)d_in[14]; const float* b_k   = (const float*)d_in[15];
    const float* W_v   = (const float*)d_in[16]; const float* b_v   = (const float*)d_in[17];
    const float* W_go  = (const float*)d_in[18]; const float* b_go  = (const float*)d_in[19];
    const float* ln_g  = (const float*)d_in[20]; const float* ln_b  = (const float*)d_in[21];
    const float* W_m1  = (const float*)d_in[22]; const float* b_m1  = (const float*)d_in[23];
    const float* W_m2  = (const float*)d_in[24]; const float* b_m2  = (const float*)d_in[25];
    const int* mask    = (const int*)d_in[26];
    const int* knn     = (const int*)d_in[27];
    const int* face_ids= (const int*)d_in[28];

    float* ws = (float*)d_ws;
    size_t off = 0;
    float* q_local    = ws + off; off += (size_t)NS * 64;
    float* q_g        = ws + off; off += (size_t)NS * 128;
    float* static_nbr = ws + off; off += (size_t)NS * 64;
    float* latent_kv  = ws + off; off += 96 * 128;
    float* kg         = ws + off; off += 96 * 128;
    float* vg         = ws + off; off += 96 * 128;
    float* dyn        = ws + off; off += (size_t)BB * NS * 64;
    float* local_feat = ws + off; off += (size_t)BB * NS * 64;
    float* gbuf       = ws + off; off += (size_t)BB * NS * 128;
    float* hbuf       = dyn;   // dyn dead after K4; reuse for MLP hidden

    k1_static<<<(NS + 15) / 16, 256, 0, stream>>>(pos_embed, face_emb, face_ids,
                                                  W_ql, b_ql, W_qg, b_qg, W_nbr, b_nbr,
                                                  q_local, q_g, static_nbr);
    k2_latent<<<BB * 6, 128, 0, stream>>>(latent_seq, W_lat, b_lat, face_emb, W_lf, b_lf, latent_kv);
    k2b_kv<<<BB * 6, 128, 0, stream>>>(latent_kv, W_k, b_k, W_v, b_v, kg, vg);
    const long tot3 = (long)BB * NS * 64;
    k3_dyn<<<(int)((tot3 + 255) / 256), 256, 0, stream>>>(x_flat, static_nbr, W_nbr, dyn);
    k4_local<<<BB * NCHUNK, 256, 0, stream>>>(dyn, q_local, knn, mask, local_feat);
    k5_global<<<BB * NCHUNK, 256, 0, stream>>>(q_g, kg, vg, gbuf);
    k6_go<<<(BB * NS) / 16, 256, 0, stream>>>(W_go, b_go, gbuf);
    k8_mlp1<<<(BB * NS) / 16, 128, 0, stream>>>(local_feat, gbuf, ln_g, ln_b, W_m1, b_m1, hbuf);
    k9_out<<<(BB * NS + 255) / 256, 256, 0, stream>>>(hbuf, W_m2, b_m2, mask, (float*)d_out);
}